// GCNEncoder_54906861912492
// MI455X (gfx1250) — compile-verified
//
#include <hip/hip_runtime.h>

#define FD 96  // feature dim (in = hidden = out)

typedef __attribute__((ext_vector_type(2))) float v2f;
typedef __attribute__((ext_vector_type(8))) float v8f;

// ---------- degree / normalization ----------
__global__ void k_deg_init(float* deg, int n) {
  int i = blockIdx.x * blockDim.x + threadIdx.x;
  if (i < n) deg[i] = 1.0f;  // self-loop contributes 1 to every dst degree
}

__global__ void k_deg_accum(const int* __restrict__ dst, float* deg, int e) {
  int i = blockIdx.x * blockDim.x + threadIdx.x;
  if (i < e) atomicAdd(&deg[dst[i]], 1.0f);
}

__global__ void k_dinv(float* deg, int n) {
  int i = blockIdx.x * blockDim.x + threadIdx.x;
  if (i < n) {
    float d = deg[i];
    deg[i] = d > 0.0f ? rsqrtf(d) : 0.0f;
  }
}

__global__ void k_zero(float* p, long long n) {
  long long i = blockIdx.x * (long long)blockDim.x + threadIdx.x;
  if (i < n) p[i] = 0.0f;
}

// ---------- dense GEMM: H[n x 96] = A[n x 96] * W[96 x 96] ----------
// One wave per 16x16 output tile; K=96 consumed by 24 V_WMMA_F32_16X16X4_F32.
// W is staged in LDS (36 KB) once per block.
__global__ __launch_bounds__(256) void k_gemm_wmma(const float* __restrict__ A,
                                                   const float* __restrict__ W,
                                                   float* __restrict__ H, int n) {
  __shared__ float wlds[FD * FD];
  for (int i = threadIdx.x; i < FD * FD; i += blockDim.x) wlds[i] = W[i];
  __syncthreads();

  const int lane = threadIdx.x & 31;
  const int wave = blockIdx.x * (blockDim.x >> 5) + (threadIdx.x >> 5);
  const int tiles_col = FD / 16;  // 6
  const int row_tile = wave / tiles_col;
  const int col_tile = wave % tiles_col;
  const int row0 = row_tile * 16;
  if (row0 >= n) return;  // wave-uniform exit (after barrier)

  const int m = lane & 15;      // row (A) / col (B,C,D) within tile
  const int half = lane >> 4;   // lane group 0|1
  const int col0 = col_tile * 16;

  int arow = row0 + m;
  if (arow >= n) arow = n - 1;  // clamp so EXEC stays all-1s for WMMA
  const float* ap = A + (long long)arow * FD;

  v8f acc = {};
#pragma unroll
  for (int k0 = 0; k0 < FD; k0 += 4) {
    const int ka = k0 + half * 2;  // A: lane half selects K pair
    v2f a, b;
    a[0] = ap[ka + 0];
    a[1] = ap[ka + 1];
    b[0] = wlds[(ka + 0) * FD + col0 + m];
    b[1] = wlds[(ka + 1) * FD + col0 + m];
    acc = __builtin_amdgcn_wmma_f32_16x16x4_f32(false, a, false, b,
                                                (short)0, acc, false, false);
  }
  // D layout: lane -> N = col0+m ; VGPR j -> M = row0 + half*8 + j
#pragma unroll
  for (int j = 0; j < 8; ++j) {
    int row = row0 + half * 8 + j;
    if (row < n) H[(long long)row * FD + col0 + m] = acc[j];
  }
}

// ---------- edge scatter-add: agg[dst] += h[src] * dinv[src]*dinv[dst] ----------
__global__ void k_edge_agg(const float* __restrict__ h,
                           const int* __restrict__ src,
                           const int* __restrict__ dst,
                           const float* __restrict__ dinv,
                           float* __restrict__ agg, int e) {
  const int CH = FD / 4;  // 24 float4 chunks per edge
  long long t = blockIdx.x * (long long)blockDim.x + threadIdx.x;
  if (t >= (long long)e * CH) return;
  int ei = (int)(t / CH);
  int c = (int)(t % CH) * 4;
  int s = src[ei], d = dst[ei];
  float nrm = dinv[s] * dinv[d];
  float4 v = *(const float4*)(h + (long long)s * FD + c);
  float* o = agg + (long long)d * FD + c;
  atomicAdd(o + 0, v.x * nrm);
  atomicAdd(o + 1, v.y * nrm);
  atomicAdd(o + 2, v.z * nrm);
  atomicAdd(o + 3, v.w * nrm);
}

// ---------- self-loop + bias (+ optional PReLU). out may alias agg. ----------
__global__ void k_post(const float* __restrict__ h, const float* agg,
                       const float* __restrict__ dinv,
                       const float* __restrict__ bias,
                       const float* prelu,  // nullptr -> no activation
                       float* out, long long n) {
  long long t = blockIdx.x * (long long)blockDim.x + threadIdx.x;
  if (t >= n * FD) return;
  int i = (int)(t / FD);
  int c = (int)(t % FD);
  float di = dinv[i];
  float v = agg[t] + h[t] * di * di + bias[c];  // self-loop norm = dinv^2
  if (prelu) {
    float a = prelu[0];
    v = v > 0.0f ? v : a * v;
  }
  out[t] = v;
}

extern "C" void kernel_launch(void* const* d_in, const int* in_sizes, int n_in,
                              void* d_out, int out_size, void* d_ws, size_t ws_size,
                              hipStream_t stream) {
  const float* x  = (const float*)d_in[0];
  const int*   ei = (const int*)d_in[1];
  const float* W1 = (const float*)d_in[2];
  const float* b1 = (const float*)d_in[3];
  const float* a1 = (const float*)d_in[4];
  const float* W2 = (const float*)d_in[5];
  const float* b2 = (const float*)d_in[6];

  const int n = in_sizes[0] / FD;
  const int e = in_sizes[1] / 2;
  const int* src = ei;
  const int* dst = ei + e;
  const long long nd = (long long)n * FD;

  // workspace: dinv[n] | h[n*96] | agg[n*96]   (~38.8 MB for N=50k)
  float* dinv = (float*)d_ws;
  float* h    = dinv + (((long long)n + 63) & ~63LL);
  float* agg  = h + nd;

  const int B = 256;
  const int waves = ((n + 15) / 16) * (FD / 16);
  const int gblocks = (waves + (B / 32) - 1) / (B / 32);
  const long long et = (long long)e * (FD / 4);

  // normalization: deg (with self loops) -> dinv = rsqrt(deg)
  k_deg_init<<<(n + B - 1) / B, B, 0, stream>>>(dinv, n);
  k_deg_accum<<<(e + B - 1) / B, B, 0, stream>>>(dst, dinv, e);
  k_dinv<<<(n + B - 1) / B, B, 0, stream>>>(dinv, n);

  // layer 1: h = x @ W1 ; aggregate ; + self loop + b1 ; PReLU  (in-place -> agg)
  k_gemm_wmma<<<gblocks, B, 0, stream>>>(x, W1, h, n);
  k_zero<<<(int)((nd + B - 1) / B), B, 0, stream>>>(agg, nd);
  k_edge_agg<<<(int)((et + B - 1) / B), B, 0, stream>>>(h, src, dst, dinv, agg, e);
  k_post<<<(int)((nd + B - 1) / B), B, 0, stream>>>(h, agg, dinv, b1, a1, agg, n);

  // layer 2: h = act @ W2 ; aggregate ; + self loop + b2 -> d_out
  k_gemm_wmma<<<gblocks, B, 0, stream>>>(agg, W2, h, n);
  k_zero<<<(int)((nd + B - 1) / B), B, 0, stream>>>(agg, nd);
  k_edge_agg<<<(int)((et + B - 1) / B), B, 0, stream>>>(h, src, dst, dinv, agg, e);
  k_post<<<(int)((nd + B - 1) / B), B, 0, stream>>>(h, agg, dinv, b2, nullptr,
                                                   (float*)d_out, n);
}